// MyGATConv_85126251807562
// MI455X (gfx1250) — compile-verified
//
#include <hip/hip_runtime.h>
#include <hip/hip_bf16.h>

// ---------------------------------------------------------------------------
// GATConv: out = segment_softmax_aggregate( leaky_relu(alpha), x@W ) + bias
//   IN_C=256, HEADS=4, OUT_C=64 -> HC = 256
//   N_SRC = N_DST = 50000, N_EDGE = 800000
// ---------------------------------------------------------------------------

#define IN_C   256
#define HC     256          // HEADS * OUT_C
#define HEADS  4
#define N_SRC  50000
#define N_DST  50000
#define NEG_SLOPE 0.2f

typedef float v2f __attribute__((ext_vector_type(2)));
typedef float v4f __attribute__((ext_vector_type(4)));
typedef float v8f __attribute__((ext_vector_type(8)));
typedef unsigned int u32x4 __attribute__((ext_vector_type(4)));
typedef int i32x4 __attribute__((ext_vector_type(4)));
typedef int i32x8 __attribute__((ext_vector_type(8)));

__device__ __forceinline__ float lrelu(float v) {
    return v >= 0.0f ? v : NEG_SLOPE * v;
}

#if __has_builtin(__builtin_amdgcn_tensor_load_to_lds) && \
    __has_builtin(__builtin_amdgcn_s_wait_tensorcnt)
#define USE_TDM 1
#else
#define USE_TDM 0
#endif

// ---------------------------------------------------------------------------
// Kernel 1: x_src = x @ W   (fp32 WMMA 16x16x4, TDM-staged B panel)
//
// Workgroup = 8 waves = 128 rows x 64 cols. Wave 0 issues a Tensor Data Mover
// 2D-tile load of the W panel [256 rows x 64 cols] into LDS (64KB), waits on
// TENSORcnt, then the barrier releases all waves. Each wave then runs the
// 16x16x4 fp32 WMMA K-loop with A fragments streamed from global (L2-resident,
// 8B/lane coalesced) and B fragments from LDS.
//
// Fragment layouts per CDNA5 ISA (wave32):
//   A 16x4 f32 : lane L -> row = L&15, VGPR0 = A[row][k0+2*(L>>4)], VGPR1 next
//   B 4x16 f32 : lane L -> col = L&15, VGPR0 = B[k0+2*(L>>4)][col], VGPR1 next
//   C/D 16x16  : lane L -> col = L&15, VGPR i = C[i + 8*(L>>4)][col]
// ---------------------------------------------------------------------------
__global__ __launch_bounds__(256)
void gat_gemm(const float* __restrict__ x,     // [N_SRC, IN_C]
              const float* __restrict__ w,     // [IN_C, HC]
              float* __restrict__ xs)          // [N_SRC, HC]
{
    __shared__ float Bs[IN_C * 64];            // 64KB: W[:, cg*64 : cg*64+64]

    const int lane = threadIdx.x & 31;
    const int wave = threadIdx.x >> 5;
    const int cg   = blockIdx.y;                         // 64-col group (0..3)
    const int rowtile = blockIdx.x * 8 + wave;           // 16-row tile id

#if USE_TDM
    if (wave == 0) {
        // ---- Tensor DMA descriptor (D#), normal mode, 2D tile ----
        // Group 0: [1:0]=count=1 | lds_addr | global_addr[56:0] | type=2
        const unsigned long long ga = (unsigned long long)(uintptr_t)(w + cg * 64);
        u32x4 g0;
        g0[0] = 1u;                                        // count=1, user mode
        g0[1] = (unsigned)(uintptr_t)(&Bs[0]);             // LDS byte address
        g0[2] = (unsigned)(ga & 0xffffffffu);
        g0[3] = (unsigned)((ga >> 32) & 0x01ffffffu) | (2u << 30);  // type=2
        // Group 1: wg_mask=0, data_size=2 (4B), no barrier/iterate/pad
        //   tensor_dim0 = 256 (row length), tensor_dim1 = 256 (num rows)
        //   tile_dim0 = 64 cols, tile_dim1 = 256 rows, tile_dim2 unused
        //   tensor_dim0_stride = 256 elements (row pitch), dim1_stride unused
        i32x8 g1;
        g1[0] = (2 << 16);                 // data_size = 4 bytes
        g1[1] = (256 << 16);               // tensor_dim0[15:0]  (=256)
        g1[2] = (256 << 16);               // dim0[31:16]=0 | tensor_dim1[15:0]
        g1[3] = (64 << 16);                // dim1[31:16]=0 | tile_dim0 = 64
        g1[4] = 256;                       // tile_dim1 = 256 | tile_dim2 = 0
        g1[5] = 256;                       // tensor_dim0_stride[31:0] = 256
        g1[6] = 0;                         // stride0[47:32]=0 | stride1[15:0]=0
        g1[7] = 0;
        i32x4 g2 = {0, 0, 0, 0};           // 2D tile: groups 2/3 unused
        i32x4 g3 = {0, 0, 0, 0};
        i32x8 g4 = {0, 0, 0, 0, 0, 0, 0, 0};   // extra group (clang-23 form)
        __builtin_amdgcn_tensor_load_to_lds(g0, g1, g2, g3, g4, 0);
        __builtin_amdgcn_s_wait_tensorcnt(0);
    }
#else
    {   // cooperative fallback copy: 4 rows x 64 cols per iteration
        const int r = threadIdx.x >> 6;    // 0..3
        const int c = threadIdx.x & 63;
        for (int r0 = 0; r0 < IN_C; r0 += 4)
            Bs[(r0 + r) * 64 + c] = w[(size_t)(r0 + r) * HC + cg * 64 + c];
    }
#endif
    __syncthreads();

    if (rowtile >= N_SRC / 16) return;     // uniform per wave; EXEC stays full

    const int half = lane >> 4;            // 0 | 1
    const int lid  = lane & 15;
    const int koff = half * 2;

    const float* arow = x + (size_t)(rowtile * 16 + lid) * IN_C + koff;

    v8f acc0 = {}, acc1 = {}, acc2 = {}, acc3 = {};

    for (int k0 = 0; k0 < IN_C; k0 += 4) {
        const v2f a = *(const v2f*)(arow + k0);
        const float* bk = &Bs[(k0 + koff) * 64 + lid];

        v2f b0; b0.x = bk[0];   b0.y = bk[0 + 64];
        v2f b1; b1.x = bk[16];  b1.y = bk[16 + 64];
        v2f b2; b2.x = bk[32];  b2.y = bk[32 + 64];
        v2f b3; b3.x = bk[48];  b3.y = bk[48 + 64];

        acc0 = __builtin_amdgcn_wmma_f32_16x16x4_f32(false, a, false, b0, (short)0, acc0, false, false);
        acc1 = __builtin_amdgcn_wmma_f32_16x16x4_f32(false, a, false, b1, (short)0, acc1, false, false);
        acc2 = __builtin_amdgcn_wmma_f32_16x16x4_f32(false, a, false, b2, (short)0, acc2, false, false);
        acc3 = __builtin_amdgcn_wmma_f32_16x16x4_f32(false, a, false, b3, (short)0, acc3, false, false);
    }

    const int rbase = rowtile * 16 + 8 * half;
#pragma unroll
    for (int i = 0; i < 8; ++i) {
        float* o = xs + (size_t)(rbase + i) * HC + cg * 64 + lid;
        o[0]  = acc0[i];
        o[16] = acc1[i];
        o[32] = acc2[i];
        o[48] = acc3[i];
    }
}

// ---------------------------------------------------------------------------
// Kernel 2: alpha_src[n][h] = <x_src[n, h*64 : h*64+64], att_src[h]>
//           alpha_dst likewise. One wave per node, shuffle reduction.
// ---------------------------------------------------------------------------
__global__ __launch_bounds__(256)
void gat_alpha(const float* __restrict__ xs,       // [N_SRC, HC]
               const float* __restrict__ att_src,  // [HEADS*64]
               const float* __restrict__ att_dst,  // [HEADS*64]
               float* __restrict__ a_src,          // [N_SRC, HEADS]
               float* __restrict__ a_dst)          // [N_SRC, HEADS]
{
    const int lane = threadIdx.x & 31;
    const int node = (blockIdx.x * blockDim.x + threadIdx.x) >> 5;
    if (node >= N_SRC) return;

    const float* xr = xs + (size_t)node * HC;
#pragma unroll
    for (int h = 0; h < HEADS; ++h) {
        const float x0 = xr[h * 64 + lane];
        const float x1 = xr[h * 64 + 32 + lane];
        float ps = x0 * att_src[h * 64 + lane] + x1 * att_src[h * 64 + 32 + lane];
        float pd = x0 * att_dst[h * 64 + lane] + x1 * att_dst[h * 64 + 32 + lane];
#pragma unroll
        for (int off = 16; off; off >>= 1) {
            ps += __shfl_xor(ps, off, 32);
            pd += __shfl_xor(pd, off, 32);
        }
        if (lane == 0) {
            a_src[node * HEADS + h] = ps;
            a_dst[node * HEADS + h] = pd;
        }
    }
}

// ---------------------------------------------------------------------------
// Kernel 3: per-dst-node segment softmax + weighted feature aggregation.
// One wave per dst node; pass 2 is a coalesced 1KB/edge gather (dominant
// cost: ~0.82 GB total, L2-resident since x_src = 51MB << 192MB L2).
// ---------------------------------------------------------------------------
__global__ __launch_bounds__(256)
void gat_aggregate(const float* __restrict__ xs,     // [N_SRC, HC]
                   const float* __restrict__ a_src,  // [N_SRC, HEADS]
                   const float* __restrict__ a_dst,  // [N_DST, HEADS]
                   const int*   __restrict__ ptr,    // [N_DST+1]
                   const int*   __restrict__ idx,    // [N_EDGE]
                   const float* __restrict__ bias,   // [HC]
                   float* __restrict__ out)          // [N_DST, HC]
{
    const int lane = threadIdx.x & 31;
    const int d    = (blockIdx.x * blockDim.x + threadIdx.x) >> 5;
    if (d >= N_DST) return;

    const int r0 = ptr[d];
    const int r1 = ptr[d + 1];

    const v4f ad = *(const v4f*)(a_dst + (size_t)d * HEADS);

    // --- pass 1a: per-head running max ---
    float m0 = -3.402823466e38f, m1 = m0, m2 = m0, m3 = m0;
    for (int e = r0 + lane; e < r1; e += 32) {
        const int s = idx[e];
        const v4f as = *(const v4f*)(a_src + (size_t)s * HEADS);
        m0 = fmaxf(m0, lrelu(as.x + ad.x));
        m1 = fmaxf(m1, lrelu(as.y + ad.y));
        m2 = fmaxf(m2, lrelu(as.z + ad.z));
        m3 = fmaxf(m3, lrelu(as.w + ad.w));
    }
#pragma unroll
    for (int off = 16; off; off >>= 1) {
        m0 = fmaxf(m0, __shfl_xor(m0, off, 32));
        m1 = fmaxf(m1, __shfl_xor(m1, off, 32));
        m2 = fmaxf(m2, __shfl_xor(m2, off, 32));
        m3 = fmaxf(m3, __shfl_xor(m3, off, 32));
    }

    // --- pass 1b: per-head exp-sum ---
    float s0 = 0.f, s1 = 0.f, s2 = 0.f, s3 = 0.f;
    for (int e = r0 + lane; e < r1; e += 32) {
        const int s = idx[e];
        const v4f as = *(const v4f*)(a_src + (size_t)s * HEADS);
        s0 += __expf(lrelu(as.x + ad.x) - m0);
        s1 += __expf(lrelu(as.y + ad.y) - m1);
        s2 += __expf(lrelu(as.z + ad.z) - m2);
        s3 += __expf(lrelu(as.w + ad.w) - m3);
    }
#pragma unroll
    for (int off = 16; off; off >>= 1) {
        s0 += __shfl_xor(s0, off, 32);
        s1 += __shfl_xor(s1, off, 32);
        s2 += __shfl_xor(s2, off, 32);
        s3 += __shfl_xor(s3, off, 32);
    }

    // --- lane-local head selection (lane owns channels [lane*8, lane*8+8)) ---
    const int cbase = lane * 8;
    const int head  = lane >> 3;   // cbase / 64
    const float mh  = head == 0 ? m0 : head == 1 ? m1 : head == 2 ? m2 : m3;
    const float sh  = head == 0 ? s0 : head == 1 ? s1 : head == 2 ? s2 : s3;
    const float adh = head == 0 ? ad.x : head == 1 ? ad.y : head == 2 ? ad.z : ad.w;
    const float inv = sh > 0.f ? 1.0f / sh : 0.f;

    // --- pass 2: weighted gather-accumulate, coalesced 1KB/edge/wave ---
    float a0 = 0.f, a1 = 0.f, a2 = 0.f, a3 = 0.f;
    float a4 = 0.f, a5 = 0.f, a6 = 0.f, a7 = 0.f;
    for (int e = r0; e < r1; ++e) {
        const int s = idx[e];
        const float eh = lrelu(a_src[(size_t)s * HEADS + head] + adh);
        const float ev = __expf(eh - mh) * inv;
        const v4f* xp = (const v4f*)(xs + (size_t)s * HC + cbase);
        const v4f v0 = xp[0];
        const v4f v1 = xp[1];
        a0 = fmaf(ev, v0.x, a0);  a1 = fmaf(ev, v0.y, a1);
        a2 = fmaf(ev, v0.z, a2);  a3 = fmaf(ev, v0.w, a3);
        a4 = fmaf(ev, v1.x, a4);  a5 = fmaf(ev, v1.y, a5);
        a6 = fmaf(ev, v1.z, a6);  a7 = fmaf(ev, v1.w, a7);
    }

    const v4f b0 = *(const v4f*)(bias + cbase);
    const v4f b1 = *(const v4f*)(bias + cbase + 4);
    v4f o0; o0.x = a0 + b0.x; o0.y = a1 + b0.y; o0.z = a2 + b0.z; o0.w = a3 + b0.w;
    v4f o1; o1.x = a4 + b1.x; o1.y = a5 + b1.y; o1.z = a6 + b1.z; o1.w = a7 + b1.w;
    v4f* op = (v4f*)(out + (size_t)d * HC + cbase);
    op[0] = o0;
    op[1] = o1;
}

// ---------------------------------------------------------------------------
// Launch: inputs in setup_inputs() order:
//   0:x 1:ptr 2:idx 3:lin_src 4:att_src 5:att_dst 6:bias 7:num_dst 8:num_src 9:num_edge
// Workspace layout (fp32): xs[N_SRC*HC] | a_src[N_SRC*H] | a_dst[N_SRC*H]
// ---------------------------------------------------------------------------
extern "C" void kernel_launch(void* const* d_in, const int* in_sizes, int n_in,
                              void* d_out, int out_size, void* d_ws, size_t ws_size,
                              hipStream_t stream) {
    const float* x       = (const float*)d_in[0];
    const int*   ptr     = (const int*)d_in[1];
    const int*   idx     = (const int*)d_in[2];
    const float* lin     = (const float*)d_in[3];
    const float* att_src = (const float*)d_in[4];
    const float* att_dst = (const float*)d_in[5];
    const float* bias    = (const float*)d_in[6];

    float* xs    = (float*)d_ws;
    float* a_src = xs + (size_t)N_SRC * HC;
    float* a_dst = a_src + (size_t)N_SRC * HEADS;
    float* out   = (float*)d_out;

    {   // GEMM: 8 row-tiles per block, 4 column groups
        const int rowblocks = (N_SRC / 16 + 7) / 8;     // 391
        gat_gemm<<<dim3(rowblocks, 4), 256, 0, stream>>>(x, lin, xs);
    }
    {   // alpha: one wave per node
        const int blocks = (N_SRC + 7) / 8;
        gat_alpha<<<blocks, 256, 0, stream>>>(xs, att_src, att_dst, a_src, a_dst);
    }
    {   // softmax + aggregate: one wave per dst node
        const int blocks = (N_DST + 7) / 8;
        gat_aggregate<<<blocks, 256, 0, stream>>>(xs, a_src, a_dst, ptr, idx, bias, out);
    }
}